// InnerOuterProductDecoder_70300024701698
// MI455X (gfx1250) — compile-verified
//
#include <hip/hip_runtime.h>
#include <hip/hip_bf16.h>

// Factorized InnerOuterProductDecoder:
//   adj[r,n,m] = u[n,r] * v[m],  u = x @ relations^T,  v = x @ outer_weight
//   out[e]     = sigmoid(u[src[e], rel[e]] * v[dst[e]])
//
// Phase A: f32 WMMA (V_WMMA_F32_16X16X4_F32) GEMM producing u [2048x16] and v [2048]
// Phase B: bandwidth-bound edge gather + sigmoid (dominant phase, ~9 MB total traffic)

typedef __attribute__((ext_vector_type(2))) float v2f;
typedef __attribute__((ext_vector_type(8))) float v8f;

#define N_NODES 2048
#define N_RELS  16
#define N_FEAT  256

__global__ __launch_bounds__(32)
void rank1_gemm_wmma(const float* __restrict__ x,
                     const float* __restrict__ relations,
                     const float* __restrict__ outer_weight,
                     float* __restrict__ u,    // [N_NODES, N_RELS]
                     float* __restrict__ v) {  // [N_NODES]
    const int tile = blockIdx.x;        // 16-row tile of x
    const int lane = threadIdx.x;       // 0..31 (wave32)
    const int half = lane >> 4;         // selects K pair within 16x4 A tile
    const int l16  = lane & 15;         // row (A) / column (B) index
    const int row  = tile * 16 + l16;

    const float* xrow = x + row * N_FEAT;
    const float* rrow = relations + l16 * N_FEAT;   // B column l16 = relation l16

    // Lane mask: outer_weight contributes only to B column 0.
    // Branch-free (no EXEC divergence in the hot loop; WMMA needs EXEC all-1s).
    const float msk = (l16 == 0) ? 1.0f : 0.0f;

    v8f cu = {};   // accumulates u tile: 16 rows x 16 relations
    v8f cv = {};   // accumulates v in column 0

    // K = 256 swept in steps of 4 -> 64 WMMA pairs per wave
    #pragma unroll 8
    for (int kk = 0; kk < N_FEAT / 4; ++kk) {
        const int kb = kk * 4 + half * 2;

        // A 16x4 f32: lane l16 holds row M=l16; half selects K={0,1} vs K={2,3}
        v2f a = *(const v2f*)(xrow + kb);
        // B 4x16 f32: lane l16 holds column N=l16 (relation index), same K split
        v2f br = *(const v2f*)(rrow + kb);
        // B for v: outer_weight broadcast-loaded by every lane, masked to column 0
        v2f ow = *(const v2f*)(outer_weight + kb);
        v2f bw; bw.x = ow.x * msk; bw.y = ow.y * msk;

        // 8-arg form: (neg_a, A, neg_b, B, c_mod, C, reuse_a, reuse_b)
        cu = __builtin_amdgcn_wmma_f32_16x16x4_f32(false, a, false, br,
                                                   (short)0, cu, false, false);
        cv = __builtin_amdgcn_wmma_f32_16x16x4_f32(false, a, false, bw,
                                                   (short)0, cv, false, false);
    }

    // C/D layout (ISA 7.12.2): VGPR j, lanes 0-15 -> M=j, lanes 16-31 -> M=j+8; N = l16
    #pragma unroll
    for (int j = 0; j < 8; ++j) {
        const int m = tile * 16 + j + 8 * half;
        u[m * N_RELS + l16] = cu[j];
        if (l16 == 0) v[m] = cv[j];       // column 0 of cv tile holds v[m]
    }
}

__global__ __launch_bounds__(256)
void edge_score_kernel(const int* __restrict__ ei, int E,
                       const float* __restrict__ u,
                       const float* __restrict__ v,
                       float* __restrict__ out) {
    const int e = blockIdx.x * blockDim.x + threadIdx.x;
    if (e >= E) return;
    const int src = ei[e];            // edge_index[0, e]
    const int rel = ei[E + e];        // edge_index[1, e]
    const int dst = ei[2 * E + e];    // edge_index[2, e]
    const float t = u[src * N_RELS + rel] * v[dst];
    out[e] = 1.0f / (1.0f + __expf(-t));
}

extern "C" void kernel_launch(void* const* d_in, const int* in_sizes, int n_in,
                              void* d_out, int out_size, void* d_ws, size_t ws_size,
                              hipStream_t stream) {
    const float* x            = (const float*)d_in[0];   // [2048, 256]
    const float* relations    = (const float*)d_in[1];   // [16, 256]
    const float* outer_weight = (const float*)d_in[2];   // [256]
    const int*   edge_index   = (const int*)d_in[3];     // [3, E] flat

    const int E = in_sizes[3] / 3;

    float* u = (float*)d_ws;                 // 2048*16 floats = 128 KB
    float* v = u + N_NODES * N_RELS;         // 2048 floats   =   8 KB

    rank1_gemm_wmma<<<N_NODES / 16, 32, 0, stream>>>(x, relations, outer_weight, u, v);

    const int blocks = (E + 255) / 256;
    edge_score_kernel<<<blocks, 256, 0, stream>>>(edge_index, E, u, v, (float*)d_out);
}